// GATLayer_3874060501520
// MI455X (gfx1250) — compile-verified
//
#include <hip/hip_runtime.h>
#include <hip/hip_bf16.h>

typedef __attribute__((ext_vector_type(2))) float v2f;
typedef __attribute__((ext_vector_type(4))) float f4;
typedef __attribute__((ext_vector_type(8))) float v8f;

#define B_   8
#define N_   512
#define FIN_ 128
#define FOUT_ 64
#define ALPHA_ 0.2f

// ---------------------------------------------------------------------------
// Kernel 1: Wh[b, n, o] = sum_k h[b, n, k] * W[k, o]  via V_WMMA_F32_16X16X4_F32
// One wave (32 threads) per 16x16 output tile. K marched in steps of 4.
// Grid: B * (N/16) * (FOUT/16) = 8*32*4 = 1024 blocks of 32 threads.
// ---------------------------------------------------------------------------
__global__ __launch_bounds__(32) void wh_wmma_kernel(
    const float* __restrict__ h, const float* __restrict__ W,
    float* __restrict__ Wh)
{
    const int tile = blockIdx.x;
    const int nt = tile & 3;            // FOUT tile (4)
    const int mt = (tile >> 2) & 31;    // N tile (32)
    const int b  = tile >> 7;           // batch (8)

    const int lane = threadIdx.x & 31;
    const int half = lane >> 4;         // 0: lanes 0-15, 1: lanes 16-31
    const int l16  = lane & 15;

    const float* A = h + (size_t)b * N_ * FIN_;   // 512 x 128 row-major
    float*       C = Wh + (size_t)b * N_ * FOUT_; // 512 x 64  row-major

    const int M0 = mt * 16;
    const int N0 = nt * 16;

    v8f c = {};
    for (int k0 = 0; k0 < FIN_; k0 += 4) {
        // A fragment 16x4: lane = M row; VGPR0/1 hold K = k0+{0,1} (lanes 0-15)
        // or K = k0+{2,3} (lanes 16-31).
        const int ka = k0 + half * 2;
        v2f a, bf;
        const float* arow = A + (size_t)(M0 + l16) * FIN_ + ka;
        a.x = arow[0];
        a.y = arow[1];
        // B fragment 4x16: VGPR v holds row K = ka+v, col N0+l16.
        bf.x = W[(size_t)(ka + 0) * FOUT_ + N0 + l16];
        bf.y = W[(size_t)(ka + 1) * FOUT_ + N0 + l16];
        c = __builtin_amdgcn_wmma_f32_16x16x4_f32(
                /*neg_a=*/false, a, /*neg_b=*/false, bf,
                /*c_mod=*/(short)0, c, /*reuse_a=*/false, /*reuse_b=*/false);
    }

    // C/D layout: VGPR v -> row M0+v (lanes 0-15) or M0+v+8 (lanes 16-31), col N0+lane%16
#pragma unroll
    for (int v = 0; v < 8; ++v) {
        C[(size_t)(M0 + v + half * 8) * FOUT_ + N0 + l16] = c[v];
    }
}

// ---------------------------------------------------------------------------
// Kernel 2: e_i[bn] = Wh[bn,:]·a[0:64],  e_j[bn] = Wh[bn,:]·a[64:128]
// ---------------------------------------------------------------------------
__global__ __launch_bounds__(256) void ei_ej_kernel(
    const float* __restrict__ Wh, const float* __restrict__ a,
    float* __restrict__ e_i, float* __restrict__ e_j)
{
    const int idx = blockIdx.x * blockDim.x + threadIdx.x;
    if (idx >= B_ * N_) return;
    const float* row = Wh + (size_t)idx * FOUT_;
    float si = 0.f, sj = 0.f;
#pragma unroll 8
    for (int o = 0; o < FOUT_; ++o) {
        const float w = row[o];
        si += w * a[o];
        sj += w * a[FOUT_ + o];
    }
    e_i[idx] = si;
    e_j[idx] = sj;
}

// ---------------------------------------------------------------------------
// Kernel 3: streaming edge-score + softmax + aggregation.
// One 256-thread block per (b, i). Each thread owns j = t and j = t + 256.
// edge_features is 536 MB (does not fit L2) -> nontemporal float4 stream.
// ---------------------------------------------------------------------------
__global__ __launch_bounds__(256) void attn_kernel(
    const float* __restrict__ ef, const float* __restrict__ U,
    const float* __restrict__ Wh, const float* __restrict__ e_i,
    const float* __restrict__ e_j, const float* __restrict__ a,
    float* __restrict__ out)
{
    __shared__ float p[2 * 256];   // softmax numerators for all 512 j
    __shared__ float red[256];     // block reduction scratch
    __shared__ float acc[256];     // 4-way partial sums for aggregation
    __shared__ float Us[FOUT_];

    const int bi = blockIdx.x;           // 0 .. 4095
    const int b  = bi >> 9;
    const int t  = threadIdx.x;

    if (t < FOUT_) Us[t] = U[t];
    __syncthreads();

    const float a_e = a[2 * FOUT_];
    const float sei = e_i[bi];
    const float* efrow = ef + (size_t)bi * N_ * FOUT_;

    // --- edge score + leaky relu for 2 j's per thread ---
    float vals[2];
#pragma unroll
    for (int r = 0; r < 2; ++r) {
        const int j = t + r * 256;
        const f4* ep = (const f4*)(efrow + (size_t)j * FOUT_);
        float dot = 0.f;
#pragma unroll
        for (int q = 0; q < FOUT_ / 4; ++q) {
            const f4 v = __builtin_nontemporal_load(&ep[q]); // 536MB stream: bypass-friendly
            const float* u = &Us[q * 4];
            dot += v.x * u[0] + v.y * u[1] + v.z * u[2] + v.w * u[3];
        }
        float e = sei + e_j[b * N_ + j] + a_e * dot;
        vals[r] = (e >= 0.f) ? e : ALPHA_ * e;
    }

    // --- row max over 512 values ---
    red[t] = fmaxf(vals[0], vals[1]);
    __syncthreads();
    for (int s = 128; s > 0; s >>= 1) {
        if (t < s) red[t] = fmaxf(red[t], red[t + s]);
        __syncthreads();
    }
    const float rowmax = red[0];
    __syncthreads();

    // --- exp + sum ---
    const float p0 = __expf(vals[0] - rowmax);
    const float p1 = __expf(vals[1] - rowmax);
    p[t] = p0;
    p[t + 256] = p1;
    red[t] = p0 + p1;
    __syncthreads();
    for (int s = 128; s > 0; s >>= 1) {
        if (t < s) red[t] += red[t + s];
        __syncthreads();
    }
    const float inv = 1.f / red[0];

    // --- aggregation: out[bi, o] = inv * sum_j p[j] * Wh[b, j, o] ---
    // 4 groups of 64 threads; group g sums j in [g*128, g*128+128).
    const int o = t & 63;
    const int g = t >> 6;
    const float* Whb = Wh + (size_t)b * N_ * FOUT_;
    float s = 0.f;
#pragma unroll 4
    for (int j = g * 128; j < g * 128 + 128; ++j) {
        s += p[j] * Whb[(size_t)j * FOUT_ + o];   // coalesced over lanes (o contiguous)
    }
    acc[t] = s;
    __syncthreads();
    if (t < 64) {
        const float r = (acc[t] + acc[t + 64] + acc[t + 128] + acc[t + 192]) * inv;
        out[(size_t)bi * FOUT_ + t] = r;
    }
}

// ---------------------------------------------------------------------------
extern "C" void kernel_launch(void* const* d_in, const int* in_sizes, int n_in,
                              void* d_out, int out_size, void* d_ws, size_t ws_size,
                              hipStream_t stream)
{
    const float* h  = (const float*)d_in[0];   // (8, 512, 128)
    const float* ef = (const float*)d_in[1];   // (8, 512, 512, 64)
    const float* W  = (const float*)d_in[2];   // (1, 128, 64)
    const float* U  = (const float*)d_in[3];   // (1, 64)
    const float* a  = (const float*)d_in[4];   // (1, 129)
    float* out = (float*)d_out;                // (8, 512, 64)

    float* Wh  = (float*)d_ws;                 // 8*512*64 floats = 1 MB
    float* e_i = Wh + (size_t)B_ * N_ * FOUT_; // 4096 floats
    float* e_j = e_i + (size_t)B_ * N_;        // 4096 floats

    // K1: Wh = h @ W via f32 WMMA. 8 * 32 * 4 tiles, one wave each.
    wh_wmma_kernel<<<dim3(B_ * (N_ / 16) * (FOUT_ / 16)), dim3(32), 0, stream>>>(h, W, Wh);

    // K2: e_i / e_j.
    ei_ej_kernel<<<dim3((B_ * N_ + 255) / 256), dim3(256), 0, stream>>>(Wh, a, e_i, e_j);

    // K3: the 536 MB streaming kernel (bandwidth floor ~23 us on MI455X).
    attn_kernel<<<dim3(B_ * N_), dim3(256), 0, stream>>>(ef, U, Wh, e_i, e_j, a, out);
}